// DAMSoftmaxLoss_74363063763042
// MI455X (gfx1250) — compile-verified
//
#include <hip/hip_runtime.h>
#include <hip/hip_bf16.h>
#include <stdint.h>

#define SCALE   15.0f
#define MARGINC 0.3f
#define LAMDAC  2.0f
#define B_ROWS  8192
#define C_COLS  10000
#define VEC4    (C_COLS / 4)     // 2500 float4 per row (exact)
#define TPB     256
#define NWAVES  (TPB / 32)
#define ITERS   10               // ceil(2500 / 256)

// One workgroup per row. Stage the 40KB row global->LDS with CDNA5 async
// copies (ASYNCcnt path), then two LDS passes: max, then exp-sum, with a
// O(1) correction for the margin-adjusted target column.
__global__ __launch_bounds__(TPB) void dam_row_kernel(
    const float* __restrict__ costh,
    const int*   __restrict__ label,
    float*       __restrict__ row_loss)
{
    __shared__ __align__(16) float srow[C_COLS];   // 40000 B staged row
    __shared__ float sred_max[NWAVES];
    __shared__ float sred_sum[NWAVES];

    const int row  = blockIdx.x;
    const int tid  = threadIdx.x;
    const int lane = tid & 31;
    const int wave = tid >> 5;

    const float* rowp = costh + (size_t)row * C_COLS;

    // ---------------- stage row: global -> LDS (async, bypasses VGPRs) ----
    {
        const uint64_t gbase = (uint64_t)(uintptr_t)rowp;
        const uint32_t lbase = (uint32_t)(uintptr_t)(&srow[0]);
#pragma unroll
        for (int it = 0; it < ITERS; ++it) {
            const int v = tid + it * TPB;          // float4 index in row
            if (v < VEC4) {
                const uint32_t goff = (uint32_t)v * 16u;   // byte offset in row
                const uint32_t loff = lbase + (uint32_t)v * 16u;
                asm volatile(
                    "global_load_async_to_lds_b128 %0, %1, %2"
                    :
                    : "v"(loff), "v"(goff), "s"(gbase)
                    : "memory");
            }
        }
        asm volatile("s_wait_asynccnt 0x0" ::: "memory");
    }
    __syncthreads();   // all waves' async copies complete -> row visible

    // ---------------- pass 1: row max of costh (m = SCALE * max) ----------
    float lmax = -1e30f;
#pragma unroll
    for (int it = 0; it < ITERS; ++it) {
        const int v = tid + it * TPB;
        if (v < VEC4) {
            const float4 x = ((const float4*)srow)[v];
            lmax = fmaxf(lmax, fmaxf(fmaxf(x.x, x.y), fmaxf(x.z, x.w)));
        }
    }
#pragma unroll
    for (int off = 16; off > 0; off >>= 1)
        lmax = fmaxf(lmax, __shfl_xor(lmax, off, 32));
    if (lane == 0) sred_max[wave] = lmax;
    __syncthreads();

    float mm = sred_max[0];
#pragma unroll
    for (int i = 1; i < NWAVES; ++i) mm = fmaxf(mm, sred_max[i]);
    const float m = SCALE * mm;   // unmodified row max of logits

    // ---------------- pass 2: sum of exp(SCALE*x - m) ----------------------
    float lsum = 0.0f;
#pragma unroll
    for (int it = 0; it < ITERS; ++it) {
        const int v = tid + it * TPB;
        if (v < VEC4) {
            const float4 x = ((const float4*)srow)[v];
            lsum += __expf(fmaf(SCALE, x.x, -m));
            lsum += __expf(fmaf(SCALE, x.y, -m));
            lsum += __expf(fmaf(SCALE, x.z, -m));
            lsum += __expf(fmaf(SCALE, x.w, -m));
        }
    }
#pragma unroll
    for (int off = 16; off > 0; off >>= 1)
        lsum += __shfl_xor(lsum, off, 32);
    if (lane == 0) sred_sum[wave] = lsum;
    __syncthreads();

    // ---------------- finalize (deterministic, fixed order) ----------------
    if (tid == 0) {
        float total = 0.0f;
#pragma unroll
        for (int i = 0; i < NWAVES; ++i) total += sred_sum[i];

        const int   lab    = label[row];
        const float cos_t  = srow[lab];
        const float t_orig = SCALE * cos_t;
        const float delta  = (MARGINC / LAMDAC) * __expf(1.0f - cos_t);
        const float t_adj  = SCALE * (cos_t - delta);

        // swap the target column's contribution for its adjusted value
        total += __expf(t_adj - m) - __expf(t_orig - m);

        const float lse = m + logf(total);
        row_loss[row]   = lse - t_adj;        // = -logp[target]
    }
}

// Deterministic fixed-order final reduction: mean over 8192 row losses.
__global__ __launch_bounds__(256) void dam_reduce_kernel(
    const float* __restrict__ row_loss,
    float*       __restrict__ out)
{
    __shared__ float sr[8];
    float s = 0.0f;
    for (int i = threadIdx.x; i < B_ROWS; i += 256)
        s += row_loss[i];
#pragma unroll
    for (int off = 16; off > 0; off >>= 1)
        s += __shfl_xor(s, off, 32);
    if ((threadIdx.x & 31) == 0) sr[threadIdx.x >> 5] = s;
    __syncthreads();
    if (threadIdx.x == 0) {
        float t = 0.0f;
#pragma unroll
        for (int i = 0; i < 8; ++i) t += sr[i];
        out[0] = t / (float)B_ROWS;
    }
}

extern "C" void kernel_launch(void* const* d_in, const int* in_sizes, int n_in,
                              void* d_out, int out_size, void* d_ws, size_t ws_size,
                              hipStream_t stream)
{
    const float* costh = (const float*)d_in[0];   // [B, C] f32
    const int*   label = (const int*)d_in[1];     // [B] int
    float* row_loss = (float*)d_ws;               // 8192 f32 = 32 KB scratch
    float* out      = (float*)d_out;              // scalar loss

    dam_row_kernel<<<B_ROWS, TPB, 0, stream>>>(costh, label, row_loss);
    dam_reduce_kernel<<<1, 256, 0, stream>>>(row_loss, out);
}